// HybridFrequencyEncoder_51024211476910
// MI455X (gfx1250) — compile-verified
//
#include <hip/hip_runtime.h>

typedef __attribute__((ext_vector_type(2))) float v2f;
typedef __attribute__((ext_vector_type(8))) float v8f;

#define W_IN   1024
#define W_OUT  512
#define PLANES 48                      // 16 * 3
#define TILES_X 32                     // 512 / 16
#define TILES_PER_PLANE (TILES_X * TILES_X)
#define N_TILES (PLANES * TILES_PER_PLANE)   // 49152
#define N_OUT   (PLANES * W_OUT * W_OUT)     // elements per output tensor
#define WAVES_PER_BLOCK 8
#define LDS_STRIDE 18                  // even stride: 8B-aligned rows, low bank conflicts

__global__ __launch_bounds__(256)
void hfe_dwt_dct_kernel(const float* __restrict__ x,
                        float* __restrict__ ll_dct,
                        float* __restrict__ lh,
                        float* __restrict__ hl,
                        float* __restrict__ hh)
{
    __shared__ float lds[WAVES_PER_BLOCK][16][LDS_STRIDE];

    const int wave = threadIdx.x >> 5;
    const int lane = threadIdx.x & 31;
    const int c = lane & 15;   // column index (N for B/C operands, row M for A operand)
    const int h = lane >> 4;   // lane-half select

    const int tile = blockIdx.x * WAVES_PER_BLOCK + wave;  // grid exactly covers N_TILES
    const int p  = tile / TILES_PER_PLANE;
    const int t  = tile % TILES_PER_PLANE;
    const int tr = t / TILES_X;
    const int tc = t % TILES_X;

    const float* in = x + (size_t)p * (W_IN * W_IN)
                        + (size_t)(tr * 32) * W_IN + tc * 32;   // 32x32 input patch

    // ---- A operand: blockdiag(D, D), D = 8x8 DCT-II matrix -----------------
    // A layout (16x4 f32): lane holds row m=c; chunk kc, VGPR v -> k = 4*kc + 2*h + v
    v2f A[4];
    #pragma unroll
    for (int kc = 0; kc < 4; ++kc) {
        #pragma unroll
        for (int v = 0; v < 2; ++v) {
            const int k = 4 * kc + 2 * h + v;
            float val = 0.0f;
            if ((c < 8) == (k < 8)) {
                const int i = c & 7, j = k & 7;
                val = (i == 0) ? 0.35355339059327373f                     // sqrt(1/8)
                               : 0.5f * __cosf(0.19634954084936207f       // pi/16
                                               * (float)(i * (2 * j + 1)));
            }
            A[kc][v] = val;
        }
    }

    // ---- Haar DWT: build LL tile in B-operand layout, write LH/HL/HH -------
    // B layout (4x16 f32): lane holds col n=c; chunk kc, VGPR v -> k = 4*kc + 2*h + v
    // The row set {4*kc + 2*h + v} partitions rows 0..15 across the wave.
    float* lhp = lh + (size_t)p * (W_OUT * W_OUT);
    float* hlp = hl + (size_t)p * (W_OUT * W_OUT);
    float* hhp = hh + (size_t)p * (W_OUT * W_OUT);
    const int col_out = tc * 16 + c;
    const int row_out0 = tr * 16;

    v2f Bm[4];
    #pragma unroll
    for (int kc = 0; kc < 4; ++kc) {
        #pragma unroll
        for (int v = 0; v < 2; ++v) {
            const int r = 4 * kc + 2 * h + v;                 // LL-local row
            const float* q = in + (size_t)(2 * r) * W_IN + 2 * c;
            const float2 t0 = *(const float2*)(q);            // x[2r][2c], x[2r][2c+1]
            const float2 t1 = *(const float2*)(q + W_IN);     // x[2r+1][2c], x[2r+1][2c+1]
            const float s00 = t0.x, s01 = t0.y, s10 = t1.x, s11 = t1.y;
            Bm[kc][v] = 0.5f * (s00 + s01 + s10 + s11);       // LL
            const size_t o = (size_t)(row_out0 + r) * W_OUT + col_out;
            lhp[o] = 0.5f * (s00 + s01 - s10 - s11);          // LH
            hlp[o] = 0.5f * (s00 - s01 + s10 - s11);          // HL
            hhp[o] = 0.5f * (s00 - s01 - s10 + s11);          // HH
        }
    }

    // ---- Pass 1: T = blockdiag(D,D) * M  (4 x K=4 chunks) ------------------
    v8f T = {0.f, 0.f, 0.f, 0.f, 0.f, 0.f, 0.f, 0.f};
    #pragma unroll
    for (int kc = 0; kc < 4; ++kc)
        T = __builtin_amdgcn_wmma_f32_16x16x4_f32(false, A[kc], false, Bm[kc],
                                                  (short)0, T, false, false);

    // ---- In-wave 16x16 transpose through LDS (C layout -> B layout of T^T) -
    // C layout: lane holds T[v + 8h][c] in VGPR v.
    #pragma unroll
    for (int v = 0; v < 8; ++v)
        lds[wave][v + 8 * h][c] = T[v];
    asm volatile("s_wait_dscnt 0" ::: "memory");   // DS in-order per wave; force completion
    v2f Bt[4];
    #pragma unroll
    for (int kc = 0; kc < 4; ++kc) {
        #pragma unroll
        for (int v = 0; v < 2; ++v)
            Bt[kc][v] = lds[wave][c][4 * kc + 2 * h + v];     // T^T[k][c] = T[c][k]
    }

    // ---- Pass 2: R^T = blockdiag(D,D) * T^T --------------------------------
    v8f R = {0.f, 0.f, 0.f, 0.f, 0.f, 0.f, 0.f, 0.f};
    #pragma unroll
    for (int kc = 0; kc < 4; ++kc)
        R = __builtin_amdgcn_wmma_f32_16x16x4_f32(false, A[kc], false, Bt[kc],
                                                  (short)0, R, false, false);

    // C layout holds R^T: lane has R[c][v + 8h] -> 8 contiguous columns per lane.
    const size_t o = (size_t)p * (W_OUT * W_OUT)
                   + (size_t)(tr * 16 + c) * W_OUT + tc * 16 + 8 * h;
    *(float4*)(ll_dct + o)     = make_float4(R[0], R[1], R[2], R[3]);
    *(float4*)(ll_dct + o + 4) = make_float4(R[4], R[5], R[6], R[7]);
}

extern "C" void kernel_launch(void* const* d_in, const int* in_sizes, int n_in,
                              void* d_out, int out_size, void* d_ws, size_t ws_size,
                              hipStream_t stream) {
    (void)in_sizes; (void)n_in; (void)out_size; (void)d_ws; (void)ws_size;
    const float* x = (const float*)d_in[0];
    float* out = (float*)d_out;                 // [LL_dct | LH | HL | HH], each N_OUT floats
    dim3 grid(N_TILES / WAVES_PER_BLOCK);       // 6144 blocks, 8 waves (tiles) each
    dim3 block(256);
    hfe_dwt_dct_kernel<<<grid, block, 0, stream>>>(
        x, out, out + (size_t)N_OUT, out + 2 * (size_t)N_OUT, out + 3 * (size_t)N_OUT);
}